// DRModel_52630529245544
// MI455X (gfx1250) — compile-verified
//
#include <hip/hip_runtime.h>
#include <hip/hip_bf16.h>
#include <math.h>

// Problem constants (match reference)
#define VOCAB 50002
#define H 256
#define B 128
#define T 100
#define K 25
#define F 266          // H + 10
#define FP 272         // F padded to multiple of 4 (K-dim of WMMA f32 16x16x4)
#define G3H 768        // 3*H

typedef __attribute__((ext_vector_type(2))) float v2f;
typedef __attribute__((ext_vector_type(8))) float v8f;

// ---------------------------------------------------------------------------
// Kernel 1: pad W_ih [768,266] -> Wp [768,272] (zero tail)
// ---------------------------------------------------------------------------
__global__ void pad_wih_kernel(const float* __restrict__ W, float* __restrict__ Wp) {
    int i = blockIdx.x * 256 + threadIdx.x;
    if (i >= G3H * FP) return;
    int n = i / FP, k = i % FP;
    Wp[i] = (k < F) ? W[n * F + k] : 0.0f;
}

// ---------------------------------------------------------------------------
// Kernel 2: build features X[b*T+t, FP]
//   cols [0,256)   masked mean-pooled embedding (padding_idx=0)
//   col 256 sin(ang), 257 cos(ang), 258 days2next/5, 259..265 one-hot dow,
//   cols 266..271 zero padding
// One block per (b,t) cell; 256 threads = one H channel each.
// ---------------------------------------------------------------------------
__global__ void build_x_kernel(const int* __restrict__ baskets,
                               const int* __restrict__ dow,
                               const int* __restrict__ hour,
                               const float* __restrict__ d2n,
                               const float* __restrict__ enc,
                               float* __restrict__ X) {
    int cell = blockIdx.x;                 // b*T + t
    int h = threadIdx.x;                   // 0..255
    const int* bk = baskets + (size_t)cell * K;
    float sum = 0.0f;
    int cnt = 0;
    #pragma unroll 5
    for (int k = 0; k < K; ++k) {
        int idx = bk[k];                   // broadcast across the block (cached)
        if (idx > 0) { sum += enc[(size_t)idx * H + h]; ++cnt; }
    }
    float* xr = X + (size_t)cell * FP;
    xr[h] = sum / (float)(cnt > 0 ? cnt : 1);
    if (h < 16) {
        float v = 0.0f;
        float ang = (float)hour[cell] / 23.0f * 3.14159265358979323846f;
        if (h == 0)       v = sinf(ang);
        else if (h == 1)  v = cosf(ang);
        else if (h == 2)  v = d2n[cell] * 0.2f;
        else if (h < 10)  v = (dow[cell] == (h - 3)) ? 1.0f : 0.0f;
        xr[256 + h] = v;                   // h in [10,16) writes the zero padding
    }
}

// ---------------------------------------------------------------------------
// Kernel 3: GI = X @ W_ih^T + b_ih  via V_WMMA_F32_16X16X4_F32
//   X  : [12800, FP]   Wp : [768, FP]   GI : [12800, 768]
// One 16x16 output tile per wave; 8 waves per 256-thread block.
// f32 WMMA layouts (ISA 7.12.2):
//   A 16x4:  lane l holds row M=l&15; vgpr{0,1} = K{0,1} (l<16) / K{2,3} (l>=16)
//   B 4x16:  lane l holds col N=l&15; vgpr{0,1} = K{0,1} (l<16) / K{2,3} (l>=16)
//   C 16x16: vgpr v holds M = v + 8*(l>=16), N = l&15
// ---------------------------------------------------------------------------
__global__ void gi_gemm_kernel(const float* __restrict__ X,
                               const float* __restrict__ Wp,
                               const float* __restrict__ b_ih,
                               float* __restrict__ GI) {
    int wid  = blockIdx.x * 8 + (threadIdx.x >> 5);   // global wave id
    int lane = threadIdx.x & 31;
    int mtile = wid / 48;                             // 800 M tiles
    int ntile = wid % 48;                             // 48 N tiles
    int l15   = lane & 15;
    int koff  = (lane >> 4) * 2;                      // 0 or 2
    int rowg  = mtile * 16 + l15;                     // A row (M)
    int colg  = ntile * 16 + l15;                     // B col (N)
    const float* xr = X  + (size_t)rowg * FP + koff;
    const float* wr = Wp + (size_t)colg * FP + koff;  // B[k,n] = W_ih[n,k]
    v8f c = {};
    for (int k = 0; k < FP; k += 4) {
        v2f a = { xr[k], xr[k + 1] };
        v2f b = { wr[k], wr[k + 1] };
        c = __builtin_amdgcn_wmma_f32_16x16x4_f32(false, a, false, b,
                                                  (short)0, c, false, false);
    }
    float bias = b_ih[colg];
    int mbase = mtile * 16 + (lane >> 4) * 8;
    #pragma unroll
    for (int v = 0; v < 8; ++v)
        GI[(size_t)(mbase + v) * G3H + colg] = c[v] + bias;
}

// ---------------------------------------------------------------------------
// Kernel 4: one GRU step (fused  GH = h @ W_hh^T + b_hh,  gates, masked write)
// Grid: 8 blocks, one per 16-row batch slab. 256 threads = 8 waves.
// Each wave computes 6 of the 48 N-tiles of GH into LDS, barrier, then
// 256 threads apply the gate nonlinearity over 16x256 hidden elements.
// ---------------------------------------------------------------------------
__global__ void gru_step_kernel(const float* __restrict__ GI,
                                const float* __restrict__ Whh,
                                const float* __restrict__ b_hh,
                                const int* __restrict__ lengths,
                                float* __restrict__ hbuf,
                                float* __restrict__ out,
                                int t) {
    __shared__ float gh_s[16 * G3H];                  // 48 KB
    int wave = threadIdx.x >> 5;
    int lane = threadIdx.x & 31;
    int rslab = blockIdx.x * 16;                      // batch rows [rslab, rslab+16)
    int l15  = lane & 15;
    int koff = (lane >> 4) * 2;
    int rowg = rslab + l15;
    const float* hr = hbuf + (size_t)rowg * H + koff;

    #pragma unroll
    for (int i = 0; i < 6; ++i) {
        int ntile = wave * 6 + i;
        int colg  = ntile * 16 + l15;
        const float* wr = Whh + (size_t)colg * H + koff;   // B[k,n] = W_hh[n,k]
        v8f c = {};
        for (int k = 0; k < H; k += 4) {
            v2f a = { hr[k], hr[k + 1] };
            v2f b = { wr[k], wr[k + 1] };
            c = __builtin_amdgcn_wmma_f32_16x16x4_f32(false, a, false, b,
                                                      (short)0, c, false, false);
        }
        float bias = b_hh[colg];
        int rb = (lane >> 4) * 8;
        #pragma unroll
        for (int v = 0; v < 8; ++v)
            gh_s[(rb + v) * G3H + colg] = c[v] + bias;
    }
    __syncthreads();

    // Gates: 16 rows x 256 cols; thread tid handles col j=tid for rows 0..15.
    int j = threadIdx.x;
    #pragma unroll 4
    for (int r = 0; r < 16; ++r) {
        int b = rslab + r;
        float hold = hbuf[(size_t)b * H + j];
        size_t gib = ((size_t)b * T + t) * G3H;
        float gr = GI[gib + j]           + gh_s[r * G3H + j];
        float gz = GI[gib + 256 + j]     + gh_s[r * G3H + 256 + j];
        float gn = GI[gib + 512 + j];
        float hn = gh_s[r * G3H + 512 + j];
        float rg = 1.0f / (1.0f + __expf(-gr));
        float zg = 1.0f / (1.0f + __expf(-gz));
        float ng = tanhf(gn + rg * hn);
        float hnew = (1.0f - zg) * ng + zg * hold;
        bool valid = t < lengths[b];
        out[((size_t)b * T + t) * H + j] = valid ? hnew : 0.0f;
        hbuf[(size_t)b * H + j] = valid ? hnew : hold;
    }
}

// ---------------------------------------------------------------------------
// Host-side launch
// ---------------------------------------------------------------------------
extern "C" void kernel_launch(void* const* d_in, const int* in_sizes, int n_in,
                              void* d_out, int out_size, void* d_ws, size_t ws_size,
                              hipStream_t stream) {
    (void)in_sizes; (void)n_in; (void)out_size; (void)ws_size;
    const int*   baskets = (const int*)  d_in[0];
    const int*   dow     = (const int*)  d_in[1];
    const int*   hour    = (const int*)  d_in[2];
    const float* d2n     = (const float*)d_in[3];
    const int*   lengths = (const int*)  d_in[4];
    const float* h0      = (const float*)d_in[5];
    const float* enc     = (const float*)d_in[6];
    const float* W_ih    = (const float*)d_in[7];
    const float* W_hh    = (const float*)d_in[8];
    const float* b_ih    = (const float*)d_in[9];
    const float* b_hh    = (const float*)d_in[10];

    float* out = (float*)d_out;                 // [B,T,H] then [1,B,H]
    float* hu_out = out + (size_t)B * T * H;

    // workspace layout (floats)
    float* ws = (float*)d_ws;
    float* X  = ws;                             // 12800 * 272
    float* Wp = X  + (size_t)(B * T) * FP;      // 768 * 272
    float* GI = Wp + (size_t)G3H * FP;          // 12800 * 768
    float* hb = GI + (size_t)(B * T) * G3H;     // 128 * 256

    // h <- h0
    hipMemcpyAsync(hb, h0, (size_t)B * H * sizeof(float),
                   hipMemcpyDeviceToDevice, stream);

    // pad W_ih
    pad_wih_kernel<<<(G3H * FP + 255) / 256, 256, 0, stream>>>(W_ih, Wp);

    // build features
    build_x_kernel<<<B * T, 256, 0, stream>>>(baskets, dow, hour, d2n, enc, X);

    // GI = X @ W_ih^T + b_ih   (800*48 tiles, 8 waves/block)
    gi_gemm_kernel<<<(800 * 48) / 8, 256, 0, stream>>>(X, Wp, b_ih, GI);

    // recurrence
    for (int t = 0; t < T; ++t)
        gru_step_kernel<<<B / 16, 256, 0, stream>>>(GI, W_hh, b_hh, lengths,
                                                    hb, out, t);

    // h_u output
    hipMemcpyAsync(hu_out, hb, (size_t)B * H * sizeof(float),
                   hipMemcpyDeviceToDevice, stream);
}